// SpectralConv2d_30777735643909
// MI455X (gfx1250) — compile-verified
//
#include <hip/hip_runtime.h>
#include <hip/hip_bf16.h>

// ---------------------------------------------------------------------------
// Truncated-spectrum FNO spectral conv for gfx1250 (MI455X), fp32 WMMA path.
//   stage A (fno_fwd_dft):  per (b,i): Xw = x * Tw  (K=256 over w), then
//                           Y = Th * Xw (K=256 over h) -> 16x16 complex modes
//   stage B (fno_mix):      per mode (k,l): complex 16x32x32 channel GEMM
//   stage C (fno_inv):      per (b,o): T = Eh * O (K=16 over k), then
//                           out = T * Bw (K=32 over re/im-l, N=256 over w)
// All twiddle operands are generated in-register with phasor recurrences.
// ---------------------------------------------------------------------------

typedef float v2f __attribute__((ext_vector_type(2)));
typedef float v8f __attribute__((ext_vector_type(8)));

#define WMMA4(A, B, C) \
  __builtin_amdgcn_wmma_f32_16x16x4_f32(false, (A), false, (B), (short)0, (C), false, false)

#define ROT(c, s, cd, sd)                         \
  do {                                            \
    float _nc = (c) * (cd) - (s) * (sd);          \
    (s) = (s) * (cd) + (c) * (sd);                \
    (c) = _nc;                                    \
  } while (0)

__device__ __forceinline__ void sincos_f(float x, float* s, float* c) { __sincosf(x, s, c); }

constexpr float kPI = 3.14159265358979323846f;

// workspace layout (floats)
#define YRE_OFF 0
#define YIM_OFF 131072
#define ORE_OFF 262144
#define OIM_OFF 393216

// ---------------------------------------------------------------------------
// Kernel 1: forward truncated DFT.  grid = 512 (b*32+i), block = 256 (8 waves)
// ---------------------------------------------------------------------------
__global__ __launch_bounds__(256) void fno_fwd_dft(const float* __restrict__ x,
                                                   float* __restrict__ ws) {
  __shared__ float xs[8][16 * 36];   // per-wave 16x32 x-chunk, stride 36 (conflict-free A reads)
  __shared__ float xw[256 * 40];     // Xw: 256 h-rows x 32 (re l | im l), stride 40

  const int tid = threadIdx.x;
  const int wave = tid >> 5, lane = tid & 31;
  const int g = lane >> 4, m = lane & 15;
  const int bi = blockIdx.x;
  const float* __restrict__ xg = x + (size_t)bi * 65536u;

  // ---- stage 1: Xw[h,l] = sum_w x[h,w] e^{-2 pi i l w / 256}, l = 0..15
  for (int t = 0; t < 2; ++t) {
    const int h0 = (wave + t * 8) * 16;
    v8f accR = {}, accI = {};
    // phasor: theta_v = 2*pi*l*(4j+2g+v)/256, l = m ; delta per chunk = pi*l/32
    float cd, sd;
    sincos_f(kPI * (float)m * (1.0f / 32.0f), &sd, &cd);
    float c0, s0, c1, s1;
    sincos_f((2.0f * kPI / 256.0f) * (float)(m * (2 * g + 0)), &s0, &c0);
    sincos_f((2.0f * kPI / 256.0f) * (float)(m * (2 * g + 1)), &s1, &c1);

    for (int kc = 0; kc < 256; kc += 32) {
      // wave-cooperative fill of the 16x32 chunk (float4, coalesced)
      for (int q = 0; q < 4; ++q) {
        int f = lane + q * 32;               // 128 float4 slots: 16 rows x 8
        int r = f >> 3, c4 = f & 7;
        float4 v = *(const float4*)(xg + (h0 + r) * 256 + kc + c4 * 4);
        *(float4*)(&xs[wave][r * 36 + c4 * 4]) = v;
      }
      if (kc + 32 < 256)
        __builtin_prefetch(xg + (h0 + (lane >> 1)) * 256 + kc + 32, 0, 3);
      asm volatile("s_wait_dscnt 0" ::: "memory");  // same-wave LDS RAW

      for (int j = 0; j < 8; ++j) {
        const int cb = j * 4 + 2 * g;
        v2f a, bc, bs;
        a.x = xs[wave][m * 36 + cb + 0];
        a.y = xs[wave][m * 36 + cb + 1];
        bc.x = c0;  bc.y = c1;    // cos(2 pi l w/256)
        bs.x = -s0; bs.y = -s1;   // -sin(...)  -> imaginary part
        accR = WMMA4(a, bc, accR);
        accI = WMMA4(a, bs, accI);
        ROT(c0, s0, cd, sd);
        ROT(c1, s1, cd, sd);
      }
    }
    for (int v = 0; v < 8; ++v) {
      const int hr = h0 + v + 8 * g;     // C layout: M = v + 8*(lane>>4), N = m
      xw[hr * 40 + m] = accR[v];
      xw[hr * 40 + 16 + m] = accI[v];
    }
  }
  __syncthreads();

  // ---- stage 2: Y[k,l] = sum_h e^{-2 pi i k h/256} Xw[h,l]; wave0 -> re, wave1 -> im
  if (wave < 2) {
    v8f acc = {};
    float cd, sd;
    sincos_f(kPI * (float)m * (1.0f / 32.0f), &sd, &cd);   // delta = 2*pi*k*4/256, k = m
    float c0, s0, c1, s1;
    sincos_f((2.0f * kPI / 256.0f) * (float)(m * (2 * g + 0)), &s0, &c0);
    sincos_f((2.0f * kPI / 256.0f) * (float)(m * (2 * g + 1)), &s1, &c1);
    const int c1off = (wave == 0) ? 0 : 16;   // Yre = cos*Xre + sin*Xim ; Yim = cos*Xim - sin*Xre
    const int c2off = 16 - c1off;
    const float sgn = (wave == 0) ? 1.0f : -1.0f;
    for (int j = 0; j < 64; ++j) {
      const int h = j * 4 + 2 * g;
      v2f ac, as_, b1, b2;
      ac.x = c0;        ac.y = c1;
      as_.x = sgn * s0; as_.y = sgn * s1;
      b1.x = xw[(h + 0) * 40 + c1off + m];
      b1.y = xw[(h + 1) * 40 + c1off + m];
      b2.x = xw[(h + 0) * 40 + c2off + m];
      b2.y = xw[(h + 1) * 40 + c2off + m];
      acc = WMMA4(ac, b1, acc);
      acc = WMMA4(as_, b2, acc);
      ROT(c0, s0, cd, sd);
      ROT(c1, s1, cd, sd);
    }
    float* __restrict__ yp = ws + ((wave == 0) ? YRE_OFF : YIM_OFF);
    const int b = bi >> 5, ci = bi & 31;
    for (int v = 0; v < 8; ++v) {
      const int k = v + 8 * g;
      yp[((k * 16 + m) * 16 + b) * 32 + ci] = acc[v];   // [mode][b][i]
    }
  }
}

// ---------------------------------------------------------------------------
// Kernel 2: per-mode complex channel mix.  grid = 256 modes, block = 128 (4 waves)
//   wave = reim*2 + ohalf:  O{re,im}[b, o(16-half)] = sum_i Y[b,i] * W[i,o]
// ---------------------------------------------------------------------------
__global__ __launch_bounds__(128) void fno_mix(const float* __restrict__ wre_g,
                                               const float* __restrict__ wim_g,
                                               float* __restrict__ ws) {
  __shared__ float ysr[16 * 34], ysi[16 * 34];   // Y[b][i], stride 34 (A-type reads)
  __shared__ float wsr[32 * 40], wsi[32 * 40];   // W[i][o], stride 40 (B-type reads)

  const int tid = threadIdx.x;
  const int wave = tid >> 5, lane = tid & 31;
  const int g = lane >> 4, m = lane & 15;
  const int mode = blockIdx.x;

  const float* __restrict__ yre = ws + YRE_OFF;
  const float* __restrict__ yim = ws + YIM_OFF;

  for (int f = tid; f < 512; f += 128) {
    int b = f >> 5, i = f & 31;
    ysr[b * 34 + i] = yre[(mode * 16 + b) * 32 + i];
    ysi[b * 34 + i] = yim[(mode * 16 + b) * 32 + i];
  }
  for (int f = tid; f < 1024; f += 128) {
    int i = f >> 5, o = f & 31;
    wsr[i * 40 + o] = wre_g[f * 256 + mode];   // w[i,o,k,l] = w[(i*32+o)*256 + mode]
    wsi[i * 40 + o] = wim_g[f * 256 + mode];
  }
  __syncthreads();

  const int reim = wave >> 1;    // 0: out_re, 1: out_im
  const int oh = wave & 1;       // o half
  v8f acc = {};
  for (int j = 0; j < 8; ++j) {
    const int i0 = j * 4 + 2 * g;
    const float yr0 = ysr[m * 34 + i0], yr1 = ysr[m * 34 + i0 + 1];
    const float yi0 = ysi[m * 34 + i0], yi1 = ysi[m * 34 + i0 + 1];
    const float wr0 = wsr[(i0 + 0) * 40 + oh * 16 + m], wr1 = wsr[(i0 + 1) * 40 + oh * 16 + m];
    const float wi0 = wsi[(i0 + 0) * 40 + oh * 16 + m], wi1 = wsi[(i0 + 1) * 40 + oh * 16 + m];
    v2f a1, a2, b1, b2;
    if (reim == 0) {   // Ore = Yre*Wre - Yim*Wim
      a1.x = yr0;  a1.y = yr1;  b1.x = wr0; b1.y = wr1;
      a2.x = -yi0; a2.y = -yi1; b2.x = wi0; b2.y = wi1;
    } else {           // Oim = Yre*Wim + Yim*Wre
      a1.x = yr0;  a1.y = yr1;  b1.x = wi0; b1.y = wi1;
      a2.x = yi0;  a2.y = yi1;  b2.x = wr0; b2.y = wr1;
    }
    acc = WMMA4(a1, b1, acc);
    acc = WMMA4(a2, b2, acc);
  }
  float* __restrict__ op = ws + ((reim == 0) ? ORE_OFF : OIM_OFF);
  for (int v = 0; v < 8; ++v) {
    const int b = v + 8 * g, o = oh * 16 + m;
    op[(mode * 16 + b) * 32 + o] = acc[v];     // [mode][b][o]
  }
}

// ---------------------------------------------------------------------------
// Kernel 3: inverse transform.  grid = 512 (b*32+o), block = 256 (8 waves)
// ---------------------------------------------------------------------------
__global__ __launch_bounds__(256) void fno_inv(const float* __restrict__ ws,
                                               float* __restrict__ out) {
  __shared__ float osr[16 * 40], osi[16 * 40];  // O[k][l] re/im, stride 40 (B-type reads)
  __shared__ float tb[8][16 * 34];              // per-wave T tile: 16 h-rows x (re l | im l)

  const int tid = threadIdx.x;
  const int wave = tid >> 5, lane = tid & 31;
  const int g = lane >> 4, m = lane & 15;
  const int bo = blockIdx.x;
  const int b = bo >> 5, o = bo & 31;
  const float* __restrict__ ore = ws + ORE_OFF;
  const float* __restrict__ oim = ws + OIM_OFF;
  float* __restrict__ og = out + (size_t)bo * 65536u;

  {
    const int k = tid >> 4, l = tid & 15;
    osr[k * 40 + l] = ore[((k * 16 + l) * 16 + b) * 32 + o];
    osi[k * 40 + l] = oim[((k * 16 + l) * 16 + b) * 32 + o];
  }
  __syncthreads();

  const float invn = 1.0f / 65536.0f;
  for (int t = 0; t < 2; ++t) {
    const int h0 = (wave + t * 8) * 16;
    const float hF = (float)(h0 + m);

    // ---- step 1: T[h,l] = sum_k e^{+2 pi i k h/256} O[k,l]   (K = 16)
    v8f aR = {}, aI = {};
    {
      float cd, sd;
      sincos_f(kPI * hF * (1.0f / 32.0f), &sd, &cd);   // delta = 2*pi*h*4/256
      float c0, s0, c1, s1;
      sincos_f((2.0f * kPI / 256.0f) * hF * (float)(2 * g + 0), &s0, &c0);
      sincos_f((2.0f * kPI / 256.0f) * hF * (float)(2 * g + 1), &s1, &c1);
      for (int j = 0; j < 4; ++j) {
        const int k0 = j * 4 + 2 * g;
        v2f ac, asp, asn, bre, bim;
        ac.x = c0;   ac.y = c1;
        asp.x = s0;  asp.y = s1;
        asn.x = -s0; asn.y = -s1;
        bre.x = osr[(k0 + 0) * 40 + m]; bre.y = osr[(k0 + 1) * 40 + m];
        bim.x = osi[(k0 + 0) * 40 + m]; bim.y = osi[(k0 + 1) * 40 + m];
        aR = WMMA4(ac, bre, aR);    // Tre = cos*Ore - sin*Oim
        aR = WMMA4(asn, bim, aR);
        aI = WMMA4(ac, bim, aI);    // Tim = cos*Oim + sin*Ore
        aI = WMMA4(asp, bre, aI);
        ROT(c0, s0, cd, sd);
        ROT(c1, s1, cd, sd);
      }
    }
    for (int v = 0; v < 8; ++v) {
      const int r = v + 8 * g;
      tb[wave][r * 34 + m] = aR[v];
      tb[wave][r * 34 + 16 + m] = aI[v];
    }
    asm volatile("s_wait_dscnt 0" ::: "memory");  // same-wave LDS RAW (tile is wave-private)

    // ---- step 2: out[h,w] = (1/65536) [ Tre[h,0] + 2*sum_{l=1..15}(Tre cos - Tim sin) ]
    for (int nt = 0; nt < 16; ++nt) {
      const int w0 = nt * 16;
      const float wF = (float)(w0 + m);
      float cd, sd;
      sincos_f(kPI * wF * (1.0f / 32.0f), &sd, &cd);   // delta = 2*pi*w*4/256 (l steps of 4)
      float ci0, si0, ci1, si1;
      sincos_f((2.0f * kPI / 256.0f) * wF * (float)(2 * g + 0), &si0, &ci0);
      sincos_f((2.0f * kPI / 256.0f) * wF * (float)(2 * g + 1), &si1, &ci1);
      float c0 = ci0, s0 = si0, c1 = ci1, s1 = si1;
      v8f acc = {};
      for (int j = 0; j < 8; ++j) {
        const int cbase = j * 4 + 2 * g;         // K index: 0..15 re(l), 16..31 im(l)
        const bool imPart = (j >= 4);
        const int l0 = cbase - (imPart ? 16 : 0);
        v2f a, bv;
        a.x = tb[wave][m * 34 + cbase + 0];
        a.y = tb[wave][m * 34 + cbase + 1];
        const float coef0 = (l0 == 0) ? invn : 2.0f * invn;
        const float coef1 = 2.0f * invn;         // l0+1 is never 0
        if (!imPart) { bv.x = coef0 * c0;    bv.y = coef1 * c1; }
        else         { bv.x = -coef0 * s0;   bv.y = -coef1 * s1; }
        acc = WMMA4(a, bv, acc);
        if (j == 3) { c0 = ci0; s0 = si0; c1 = ci1; s1 = si1; }  // restart l for im pass
        else        { ROT(c0, s0, cd, sd); ROT(c1, s1, cd, sd); }
      }
      for (int v = 0; v < 8; ++v)
        og[(h0 + v + 8 * g) * 256 + w0 + m] = acc[v];
    }
  }
}

// ---------------------------------------------------------------------------
extern "C" void kernel_launch(void* const* d_in, const int* in_sizes, int n_in,
                              void* d_out, int out_size, void* d_ws, size_t ws_size,
                              hipStream_t stream) {
  const float* x = (const float*)d_in[0];
  const float* w_real = (const float*)d_in[1];
  const float* w_imag = (const float*)d_in[2];
  float* out = (float*)d_out;
  float* ws = (float*)d_ws;   // uses 2 MB: Y (1 MB) + O (1 MB)

  fno_fwd_dft<<<512, 256, 0, stream>>>(x, ws);
  fno_mix<<<256, 128, 0, stream>>>(w_real, w_imag, ws);
  fno_inv<<<512, 256, 0, stream>>>(ws, out);
}